// AttentionHead_2869038154391
// MI455X (gfx1250) — compile-verified
//
#include <hip/hip_runtime.h>

#define WINDOW   32
#define DIM      4096
#define SCALE_F  (1.0f / 256.0f)
#define MASK_MAG 100000.0f
#define KC       32      // K-chunk per LDS stage
#define NT       32      // output columns per block (2 waves x 16)
#define XPAD     4       // +4 floats: keeps rows 16B-aligned for async b128,
                         // stride 36 dwords -> conflict-free lane access
#define BLK      64      // threads per block (2 waves)

typedef __attribute__((ext_vector_type(2))) float v2f;
typedef __attribute__((ext_vector_type(8))) float v8f;

__device__ __forceinline__ v8f wmma_f32_16x16x4(v2f a, v2f b, v8f c) {
    // 8 args: (neg_a, A, neg_b, B, c_mod, C, reuse_a, reuse_b)
    return __builtin_amdgcn_wmma_f32_16x16x4_f32(false, a, false, b,
                                                 (short)0, c, false, false);
}

// Async global -> LDS copy, 16B per lane, tracked by ASYNCcnt (CDNA5 §10/§15.18).
__device__ __forceinline__ void async_copy_b128(const float* gaddr, const void* lds) {
    const unsigned loff = (unsigned)(unsigned long long)lds;  // low 32b = LDS offset
    asm volatile("global_load_async_to_lds_b128 %0, %1, off"
                 :: "v"(loff), "v"(gaddr) : "memory");
}

// ---------------------------------------------------------------------------
// Kernel 1: fused K/Q/V GEMM.  out[m][n] = sum_k x[m][k] * W[k][n]
// grid.x = 3 * (DIM/NT); block = 64 (2 waves, each wave owns 16 columns,
// two 16x16 C tiles stacked in M).  Weights + x stream via async-to-LDS with
// double buffering: issue chunk i+1, s_wait_asynccnt 8 (8 newest = chunk i+1
// still pending -> chunk i landed), barrier, WMMA on chunk i.
// ---------------------------------------------------------------------------
__global__ __launch_bounds__(BLK) void qkv_gemm(const float* __restrict__ x,
                                                const float* __restrict__ wk,
                                                const float* __restrict__ wq,
                                                const float* __restrict__ wvp,
                                                float* __restrict__ ws) {
    __shared__ __align__(16) float xs[2][WINDOW][KC + XPAD];
    __shared__ __align__(16) float wl[2][KC][NT + XPAD];

    const int tiles_per_mat = DIM / NT;
    const int mat  = blockIdx.x / tiles_per_mat;
    const int tile = blockIdx.x % tiles_per_mat;
    const int n0b  = tile * NT;
    const float* W = (mat == 0) ? wk : (mat == 1) ? wq : wvp;
    float* Out = ws + (size_t)mat * WINDOW * DIM;

    const int tid  = threadIdx.x;
    const int wid  = tid >> 5;
    const int lane = tid & 31;
    const int l    = lane & 15;
    const int hi   = lane >> 4;            // 0: K=0,1 / M=r   1: K=2,3 / M=r+8
    const int cw   = wid * 16 + l;         // column within block tile
    const int cn   = n0b + cw;             // global output column

    // Issue the 8 async b128 copies (4 for W, 4 for x) for one K-chunk.
    auto issue_chunk = [&](int k0, int b) {
        #pragma unroll
        for (int i = tid; i < (KC * NT) / 4; i += BLK) {       // 4 rounds
            const int row = i >> 3;                            // NT/4 = 8 f4/row
            const int c4  = i & 7;
            async_copy_b128(W + (size_t)(k0 + row) * DIM + n0b + c4 * 4,
                            &wl[b][row][c4 * 4]);
        }
        #pragma unroll
        for (int i = tid; i < (WINDOW * KC) / 4; i += BLK) {   // 4 rounds
            const int row = i >> 3;                            // KC/4 = 8 f4/row
            const int c4  = i & 7;
            async_copy_b128(x + (size_t)row * DIM + k0 + c4 * 4,
                            &xs[b][row][c4 * 4]);
        }
    };

    v8f c0 = {}; v8f c1 = {};

    issue_chunk(0, 0);
    int buf = 0;
    for (int k0 = 0; k0 < DIM; k0 += KC) {
        if (k0 + KC < DIM) {
            issue_chunk(k0 + KC, buf ^ 1);
            asm volatile("s_wait_asynccnt 0x8" ::: "memory");  // chunk k0 landed
        } else {
            asm volatile("s_wait_asynccnt 0x0" ::: "memory");
        }
        __syncthreads();

        const float (*xc)[KC + XPAD] = xs[buf];
        const float (*wc)[NT + XPAD] = wl[buf];
        #pragma unroll
        for (int s = 0; s < KC / 4; ++s) {
            const int kk = 4 * s + 2 * hi;
            const v2f a0 = *(const v2f*)&xc[l][kk];        // M = 0..15 tile
            const v2f a1 = *(const v2f*)&xc[16 + l][kk];   // M = 16..31 tile
            v2f b;
            b.x = wc[kk][cw];
            b.y = wc[kk + 1][cw];
            c0 = wmma_f32_16x16x4(a0, b, c0);
            c1 = wmma_f32_16x16x4(a1, b, c1);
        }
        __syncthreads();
        buf ^= 1;
    }

    #pragma unroll
    for (int r = 0; r < 8; ++r) {
        Out[(size_t)(r + 8 * hi) * DIM + cn]      = c0[r];
        Out[(size_t)(16 + r + 8 * hi) * DIM + cn] = c1[r];
    }
}

// ---------------------------------------------------------------------------
// Kernel 2: S = (Q K^T) * SCALE, min-mask, row softmax -> attn (32x32).
// One block, 4 waves; wave w computes 16x16 tile (mi,ni)=(w>>1, w&1).
// ---------------------------------------------------------------------------
__global__ __launch_bounds__(128) void scores_softmax(const float* __restrict__ ws,
                                                      float* __restrict__ attn) {
    const float* kmat = ws;                          // K at offset 0
    const float* qmat = ws + (size_t)WINDOW * DIM;   // Q at offset N
    __shared__ float s[WINDOW][WINDOW + 2];

    const int tid  = threadIdx.x;
    const int wid  = tid >> 5;
    const int lane = tid & 31;
    const int l    = lane & 15;
    const int hi   = lane >> 4;
    const int m0   = (wid >> 1) * 16;
    const int n0   = (wid & 1) * 16;

    v8f c = {};
    for (int kk = 0; kk < DIM; kk += 4) {
        // A = Q rows m0..m0+15; B = K^T -> rows of K across lanes (float2 = K,K+1)
        const v2f a = *(const v2f*)(qmat + (size_t)(m0 + l) * DIM + kk + 2 * hi);
        const v2f b = *(const v2f*)(kmat + (size_t)(n0 + l) * DIM + kk + 2 * hi);
        c = wmma_f32_16x16x4(a, b, c);
    }
    #pragma unroll
    for (int r = 0; r < 8; ++r)
        s[m0 + r + 8 * hi][n0 + l] = c[r] * SCALE_F;
    __syncthreads();

    if (tid < WINDOW) {
        float row[WINDOW];
        float mx = -3.0e38f;
        for (int j = 0; j < WINDOW; ++j) {
            const float msk = (j <= tid) ? MASK_MAG : -MASK_MAG;
            const float v = fminf(s[tid][j], msk);
            row[j] = v;
            mx = fmaxf(mx, v);
        }
        float sum = 0.0f;
        for (int j = 0; j < WINDOW; ++j) {
            row[j] = expf(row[j] - mx);
            sum += row[j];
        }
        const float inv = 1.0f / sum;
        for (int j = 0; j < WINDOW; ++j)
            attn[tid * WINDOW + j] = row[j] * inv;
    }
}

// ---------------------------------------------------------------------------
// Kernel 3: out = attn (32x32) @ V (32xDIM).  grid.x = DIM/64, block = 128.
// ---------------------------------------------------------------------------
__global__ __launch_bounds__(128) void av_gemm(const float* __restrict__ vmat,
                                               const float* __restrict__ attn,
                                               float* __restrict__ out) {
    __shared__ float as[WINDOW][WINDOW + 2];
    const int tid = threadIdx.x;
    for (int i = tid; i < WINDOW * WINDOW; i += 128)
        as[i >> 5][i & 31] = attn[i];
    __syncthreads();

    const int wid  = tid >> 5;
    const int lane = tid & 31;
    const int l    = lane & 15;
    const int hi   = lane >> 4;
    const int cn   = blockIdx.x * 64 + wid * 16 + l;

    v8f c0 = {}; v8f c1 = {};
    #pragma unroll
    for (int s = 0; s < WINDOW / 4; ++s) {
        const int kk = 4 * s + 2 * hi;
        const v2f a0 = *(const v2f*)&as[l][kk];
        const v2f a1 = *(const v2f*)&as[16 + l][kk];
        v2f b;
        b.x = vmat[(size_t)kk * DIM + cn];
        b.y = vmat[(size_t)(kk + 1) * DIM + cn];
        c0 = wmma_f32_16x16x4(a0, b, c0);
        c1 = wmma_f32_16x16x4(a1, b, c1);
    }
    #pragma unroll
    for (int r = 0; r < 8; ++r) {
        out[(size_t)(r + 8 * hi) * DIM + cn]      = c0[r];
        out[(size_t)(16 + r + 8 * hi) * DIM + cn] = c1[r];
    }
}

extern "C" void kernel_launch(void* const* d_in, const int* in_sizes, int n_in,
                              void* d_out, int out_size, void* d_ws, size_t ws_size,
                              hipStream_t stream) {
    (void)in_sizes; (void)n_in; (void)out_size; (void)ws_size;
    const float* x   = (const float*)d_in[0];
    const float* wk  = (const float*)d_in[1];
    const float* wq  = (const float*)d_in[2];
    const float* wvp = (const float*)d_in[3];
    float* ws   = (float*)d_ws;                          // K | Q | V | attn
    float* attn = ws + (size_t)3 * WINDOW * DIM;
    float* out  = (float*)d_out;

    qkv_gemm<<<3 * (DIM / NT), BLK, 0, stream>>>(x, wk, wq, wvp, ws);
    scores_softmax<<<1, 128, 0, stream>>>(ws, attn);
    av_gemm<<<DIM / 64, 128, 0, stream>>>(ws + (size_t)2 * WINDOW * DIM, attn, out);
}